// GNNLayer_8658654069051
// MI455X (gfx1250) — compile-verified
//
#include <hip/hip_runtime.h>
#include <hip/hip_bf16.h>

typedef __attribute__((ext_vector_type(2))) float v2f;
typedef __attribute__((ext_vector_type(8))) float v8f;

#define IN_F  128
#define OUT_F 64

// ---------------------------------------------------------------------------
// Phase 0: zero the output accumulator (atomics accumulate into it).
// ---------------------------------------------------------------------------
__global__ void gnn_zero_kernel(float* __restrict__ out, int n) {
    int i = blockIdx.x * blockDim.x + threadIdx.x;
    if (i < n) out[i] = 0.0f;
}

// ---------------------------------------------------------------------------
// Phase 1: support = X @ W via V_WMMA_F32_16X16X4_F32.
// One wave computes a 16x16 tile of support. Block = 128 threads = 4 waves,
// covering the 4 column tiles (OUT_F = 64 = 4*16) of one 16-row stripe.
// K = 128 -> 32 WMMA issues per tile, accumulator held in 8 VGPRs.
// ---------------------------------------------------------------------------
__global__ __launch_bounds__(128) void gnn_gemm_wmma_kernel(
    const float* __restrict__ X,   // [nrows][128]
    const float* __restrict__ W,   // [128][64]
    float* __restrict__ S,         // [nrows][64]
    int nrows)
{
    const int lane = threadIdx.x & 31;
    const int wv   = threadIdx.x >> 5;          // 0..3 -> column tile
    const int row0 = blockIdx.x * 16;
    const int col0 = wv * 16;
    if (row0 + 16 > nrows) return;              // full tiles only (grid is exact)

    const int mn  = lane & 15;                  // A row / B,D col within tile
    const int khi = (lane >> 4) << 1;           // lanes 0-15: K+0,K+1; 16-31: K+2,K+3

    v8f acc = {};
    const float* __restrict__ arow = X + (size_t)(row0 + mn) * IN_F;
    const float* __restrict__ bcol = W + col0 + mn;

#pragma unroll
    for (int k = 0; k < IN_F; k += 4) {
        v2f a, b;
        a.x = arow[k + khi];
        a.y = arow[k + khi + 1];
        b.x = bcol[(size_t)(k + khi) * OUT_F];
        b.y = bcol[(size_t)(k + khi + 1) * OUT_F];
        // 8 args: (neg_a, A, neg_b, B, c_mod, C, reuse_a, reuse_b)
        acc = __builtin_amdgcn_wmma_f32_16x16x4_f32(
            false, a, false, b, (short)0, acc, false, false);
    }

    // D layout: lane holds col mn; VGPR i = row ((lane>>4)*8)+i.
    const int rbase = row0 + ((lane >> 4) << 3);
#pragma unroll
    for (int i = 0; i < 8; ++i)
        S[(size_t)(rbase + i) * OUT_F + col0 + mn] = acc[i];
}

// Scalar cleanup for a row tail (launches 0 blocks when nrows % 16 == 0).
__global__ void gnn_gemm_tail_kernel(
    const float* __restrict__ X, const float* __restrict__ W,
    float* __restrict__ S, int row_start, int nrows)
{
    int idx = blockIdx.x * blockDim.x + threadIdx.x;
    int row = row_start + idx / OUT_F;
    int col = idx % OUT_F;
    if (row >= nrows) return;
    float acc = 0.0f;
    const float* xr = X + (size_t)row * IN_F;
#pragma unroll 8
    for (int k = 0; k < IN_F; ++k)
        acc += xr[k] * W[(size_t)k * OUT_F + col];
    S[(size_t)row * OUT_F + col] = acc;
}

// ---------------------------------------------------------------------------
// Phase 2: edge scatter. One wave per edge: gather support[src] (coalesced
// 256B), scale by edge_weight, atomic-add into out[dst]. support fits in the
// 192MB L2, so both the gather and the f32 atomics stay on-chip.
// ---------------------------------------------------------------------------
__global__ __launch_bounds__(256) void gnn_scatter_kernel(
    const float* __restrict__ S,
    const int*   __restrict__ edge_index,  // [2][E]: row 0 = dst, row 1 = src
    const float* __restrict__ ew,
    float* __restrict__ out,
    int E)
{
    const int e    = (int)((blockIdx.x * 256u + threadIdx.x) >> 5);
    const int lane = threadIdx.x & 31;
    if (e >= E) return;

    const int   dst = edge_index[e];
    const int   src = edge_index[E + e];
    const float w   = ew[e];

    const float2 s = *(const float2*)(S + (size_t)src * OUT_F + lane * 2);
    float* o = out + (size_t)dst * OUT_F + lane * 2;
    atomicAdd(o,     s.x * w);   // global_atomic_add_f32, no-return
    atomicAdd(o + 1, s.y * w);
}

// ---------------------------------------------------------------------------
// Phase 3: ReLU in place.
// ---------------------------------------------------------------------------
__global__ void gnn_relu_kernel(float* __restrict__ out, int n) {
    int i = blockIdx.x * blockDim.x + threadIdx.x;
    if (i < n) {
        float v = out[i];
        out[i] = v > 0.0f ? v : 0.0f;
    }
}

extern "C" void kernel_launch(void* const* d_in, const int* in_sizes, int n_in,
                              void* d_out, int out_size, void* d_ws, size_t ws_size,
                              hipStream_t stream)
{
    const float* features    = (const float*)d_in[0];  // [N][128]
    const float* weight      = (const float*)d_in[1];  // [128][64]
    const int*   edge_index  = (const int*)  d_in[2];  // [2][E]
    const float* edge_weight = (const float*)d_in[3];  // [E]
    float*       out         = (float*)d_out;          // [N][64]
    float*       support     = (float*)d_ws;           // [N][64] scratch

    const int N = in_sizes[0] / IN_F;
    const int E = in_sizes[3];
    const int out_elems = N * OUT_F;

    // 0: zero accumulator
    gnn_zero_kernel<<<(out_elems + 255) / 256, 256, 0, stream>>>(out, out_elems);

    // 1: support = X @ W (WMMA fp32)
    const int full_tiles = N / 16;
    if (full_tiles > 0)
        gnn_gemm_wmma_kernel<<<full_tiles, 128, 0, stream>>>(features, weight,
                                                             support, N);
    const int tail_rows = N - full_tiles * 16;
    if (tail_rows > 0) {
        const int tail_threads = tail_rows * OUT_F;
        gnn_gemm_tail_kernel<<<(tail_threads + 127) / 128, 128, 0, stream>>>(
            features, weight, support, full_tiles * 16, N);
    }

    // 2: edge scatter with f32 atomics (one wave per edge)
    const long long total_scatter_threads = (long long)E * 32;
    const int scatter_blocks = (int)((total_scatter_threads + 255) / 256);
    gnn_scatter_kernel<<<scatter_blocks, 256, 0, stream>>>(support, edge_index,
                                                           edge_weight, out, E);

    // 3: ReLU
    gnn_relu_kernel<<<(out_elems + 255) / 256, 256, 0, stream>>>(out, out_elems);
}